// TtMambaSSM_66030827209148
// MI455X (gfx1250) — compile-verified
//
#include <hip/hip_runtime.h>

typedef __attribute__((ext_vector_type(2))) float v2f;
typedef __attribute__((ext_vector_type(8))) float v8f;

#define B_USERS 32
#define D_INNER 5120
#define DT_RANK 160
#define N_STATE 32
#define KSPLIT  8

static __device__ __forceinline__ v8f wmma_f32_16x16x4(v2f a, v2f b, v8f c) {
  // (neg_a, A, neg_b, B, c_mod, C, reuse_a, reuse_b) -> v_wmma_f32_16x16x4_f32
  return __builtin_amdgcn_wmma_f32_16x16x4_f32(false, a, false, b, (short)0, c,
                                               false, false);
}

// ---------------------------------------------------------------------------
// K1: split-K WMMA GEMMs:  t = xb@W_dt1 (32x160), Bm = xb@W_B, Cm = xb@W_C.
// 28 output tiles (20 + 4 + 4) x 8 K-chunks = 224 waves; f32 atomicAdd reduce.
// ---------------------------------------------------------------------------
__global__ void mamba_proj_splitk(const float* __restrict__ xb,
                                  const float* __restrict__ Wdt1,
                                  const float* __restrict__ WB,
                                  const float* __restrict__ WC,
                                  float* __restrict__ t_out,
                                  float* __restrict__ Bm,
                                  float* __restrict__ Cm) {
  const int wave = blockIdx.x * (blockDim.x >> 5) + (threadIdx.x >> 5);
  const int lane = threadIdx.x & 31;
  const int tile = wave / KSPLIT;
  const int kc   = wave % KSPLIT;
  if (tile >= 28) return;
  const int half = lane >> 4;
  const int r    = lane & 15;

  const float* W; float* out; int N, tt;
  if (tile < 20)      { W = Wdt1; out = t_out; N = DT_RANK; tt = tile;      }
  else if (tile < 24) { W = WB;   out = Bm;    N = N_STATE; tt = tile - 20; }
  else                { W = WC;   out = Cm;    N = N_STATE; tt = tile - 24; }
  const int m0 = (tt & 1) * 16;   // M = 32 -> 2 row tiles
  const int n0 = (tt >> 1) * 16;

  const int kchunk = D_INNER / KSPLIT;
  const int kbeg   = kc * kchunk;
  const float* arow = xb + (size_t)(m0 + r) * D_INNER;

  v8f c = {};
  for (int k0 = kbeg; k0 < kbeg + kchunk; k0 += 4) {
    const int ka = k0 + (half << 1);
    v2f a, b;
    a.x = arow[ka];                    // A[r][2*half+0]
    a.y = arow[ka + 1];                // A[r][2*half+1]
    b.x = W[(size_t)ka * N + n0 + r];  // B[2*half+0][r]
    b.y = W[(size_t)(ka + 1) * N + n0 + r];
    c = wmma_f32_16x16x4(a, b, c);
  }
#pragma unroll
  for (int j = 0; j < 8; ++j)          // c[j] = D[m0 + j + 8*half][n0 + r]
    atomicAdd(&out[(size_t)(m0 + j + 8 * half) * N + n0 + r], c[j]);
}

// ---------------------------------------------------------------------------
// K2: dt = softplus(t @ W_dt2 + b_dt).  M=32, N=5120, K=160.
// 2 x 320 = 640 wave-tiles, 40 WMMA steps each, fused bias+softplus.
// ---------------------------------------------------------------------------
__global__ void mamba_dtproj(const float* __restrict__ t_in,   // 32 x 160
                             const float* __restrict__ Wdt2,   // 160 x 5120
                             const float* __restrict__ b_dt,   // 5120
                             float* __restrict__ dt) {         // 32 x 5120
  const int wave = blockIdx.x * (blockDim.x >> 5) + (threadIdx.x >> 5);
  const int lane = threadIdx.x & 31;
  const int half = lane >> 4;
  const int r    = lane & 15;
  const int m0 = (wave & 1) * 16;
  const int n0 = (wave >> 1) * 16;     // 0..319 -> n0 up to 5104

  v8f c = {};
  for (int k0 = 0; k0 < DT_RANK; k0 += 4) {
    const int ka = k0 + (half << 1);
    v2f a, b;
    a.x = t_in[(m0 + r) * DT_RANK + ka];
    a.y = t_in[(m0 + r) * DT_RANK + ka + 1];
    b.x = Wdt2[(size_t)ka * D_INNER + n0 + r];
    b.y = Wdt2[(size_t)(ka + 1) * D_INNER + n0 + r];
    c = wmma_f32_16x16x4(a, b, c);
  }
  const float bias = b_dt[n0 + r];
#pragma unroll
  for (int j = 0; j < 8; ++j) {
    float v = c[j] + bias;
    // softplus(beta=1, threshold=20)
    float sp = (v > 20.f) ? v : __logf(1.f + __expf(v));
    dt[(size_t)(m0 + j + 8 * half) * D_INNER + n0 + r] = sp;
  }
}

// ---------------------------------------------------------------------------
// K3: fused selective-scan step + output projection. One wave per (b,h);
// 32 lanes = 32 state entries n (wave32-exact). Single pass over h0 (21 MB,
// the dominant traffic). Butterfly __shfl_xor reduction over the wave.
//   y[b,h] = sum_n (exp(-dt*exp(A_log[h,n]))*h0 + dt*Bm[b,n]*xb) * Cm[b,n]
//            + xb * D[h]
// ---------------------------------------------------------------------------
__global__ void mamba_scan(const float* __restrict__ xb,     // 32 x 5120
                           const float* __restrict__ dt,     // 32 x 5120
                           const float* __restrict__ A_log,  // 5120 x 32
                           const float* __restrict__ Bm,     // 32 x 32
                           const float* __restrict__ Cm,     // 32 x 32
                           const float* __restrict__ Dv,     // 5120
                           const float* __restrict__ h0,     // 32 x 5120 x 32
                           float* __restrict__ y) {          // 32 x 5120
  const int warp = threadIdx.x >> 5;
  const int lane = threadIdx.x & 31;
  const int h = blockIdx.x * (blockDim.x >> 5) + warp;
  const int b = blockIdx.y;

  const float dtv = dt[(size_t)b * D_INNER + h];
  const float xv  = xb[(size_t)b * D_INNER + h];
  const float al  = A_log[(size_t)h * N_STATE + lane];
  const float hv  = h0[((size_t)b * D_INNER + h) * N_STATE + lane];
  const float Bn  = Bm[b * N_STATE + lane];
  const float Cn  = Cm[b * N_STATE + lane];

  const float abar = __expf(-dtv * __expf(al));     // exp(dt * A), A=-exp(A_log)
  float val = (abar * hv + dtv * Bn * xv) * Cn;

#pragma unroll
  for (int off = 16; off >= 1; off >>= 1)
    val += __shfl_xor(val, off, 32);

  if (lane == 0)
    y[(size_t)b * D_INNER + h] = val + xv * Dv[h];
}

// ---------------------------------------------------------------------------
extern "C" void kernel_launch(void* const* d_in, const int* in_sizes, int n_in,
                              void* d_out, int out_size, void* d_ws, size_t ws_size,
                              hipStream_t stream) {
  const float* x     = (const float*)d_in[0];  // (1,1,32,5120)
  const float* W_dt1 = (const float*)d_in[1];  // (5120,160)
  const float* W_dt2 = (const float*)d_in[2];  // (160,5120)
  const float* b_dt  = (const float*)d_in[3];  // (5120,)
  const float* W_B   = (const float*)d_in[4];  // (5120,32)
  const float* W_C   = (const float*)d_in[5];  // (5120,32)
  const float* A_log = (const float*)d_in[6];  // (5120,32)
  const float* Dv    = (const float*)d_in[7];  // (5120,)
  const float* h0    = (const float*)d_in[8];  // (32,5120,32)
  float* out = (float*)d_out;

  float* ws    = (float*)d_ws;
  float* ws_t  = ws;                               // 32*160
  float* ws_B  = ws_t + B_USERS * DT_RANK;         // 32*32
  float* ws_C  = ws_B + B_USERS * N_STATE;         // 32*32
  float* ws_dt = ws_C + B_USERS * N_STATE;         // 32*5120

  // zero the split-K accumulators (t, Bm, Cm) -- capturable memset node
  hipMemsetAsync(d_ws, 0,
                 (B_USERS * DT_RANK + 2 * B_USERS * N_STATE) * sizeof(float),
                 stream);

  // K1: 28 tiles * 8 K-chunks = 224 waves -> 56 blocks of 4 waves
  mamba_proj_splitk<<<56, 128, 0, stream>>>(x, W_dt1, W_B, W_C,
                                            ws_t, ws_B, ws_C);

  // K2: 640 wave-tiles -> 160 blocks of 4 waves
  mamba_dtproj<<<160, 128, 0, stream>>>(ws_t, W_dt2, b_dt, ws_dt);

  // K3: one wave per (b,h): grid (5120/8, 32), 8 waves per block
  dim3 g3(D_INNER / 8, B_USERS);
  mamba_scan<<<g3, 256, 0, stream>>>(x, ws_dt, A_log, ws_B, ws_C, Dv, h0, out);
}